// AttnDecoderRNN_65446711656704
// MI455X (gfx1250) — compile-verified
//
#include <hip/hip_runtime.h>
#include <math.h>

// Sizes from the reference
#define V_SZ 50257
#define E_SZ 512
#define H_SZ 1024
#define L_SZ 512
#define EH_SZ (E_SZ + H_SZ)   // 1536

typedef float v2f __attribute__((ext_vector_type(2)));
typedef float v8f __attribute__((ext_vector_type(8)));

// ---------------------------------------------------------------------------
// Workspace layout (float offsets)
// ---------------------------------------------------------------------------
#define OFF_X1      0              // [1536] cat(embedded, h0)
#define OFF_SCORES  1536           // [512]  attn logits
#define OFF_ATTNW   2048           // [512]  attn weights
#define OFF_X2      2560           // [1536] cat(embedded, attn_applied)
#define OFF_PRERELU 4096           // [1024] combine pre-activation
#define OFF_X3      5120           // [1024] relu output (GRU input)
#define OFF_GI      6144           // [3072]
#define OFF_GH      9216           // [3072]
#define OFF_H1      12288          // [1024]
#define OFF_LOGITS  13312          // [50257]
#define OFF_RED     63569          // [1] logsumexp scalar

// ---------------------------------------------------------------------------
// Generic WMMA-based GEMV:  y[r] = dot(W[r,:], x) + bias[r]
// One wave computes a 16-row tile using V_WMMA_F32_16X16X4_F32.
// A fragment: 16x4 fp32 tile of W (lane m=lane&15 holds row row0+m,
//             K pair selected by hi=lane>>4). B fragment: x chunk broadcast
//             into all 16 columns, so every column of D equals W·x.
// ---------------------------------------------------------------------------
__global__ void wmma_gemv(const float* __restrict__ W,
                          const float* __restrict__ x,
                          const float* __restrict__ bias,
                          float* __restrict__ y,
                          int rows, int cols) {
    extern __shared__ float sx[];   // cols floats
    for (int i = threadIdx.x; i < cols; i += blockDim.x) sx[i] = x[i];
    __syncthreads();

    const int wave = threadIdx.x >> 5;
    const int lane = threadIdx.x & 31;
    const int tile = blockIdx.x * (blockDim.x >> 5) + wave;
    const int row0 = tile * 16;
    if (row0 >= rows) return;     // uniform per wave: EXEC stays all-ones

    const int m  = lane & 15;
    const int hi = lane >> 4;     // 0: K={k,k+1}, 1: K={k+2,k+3}
    int row = row0 + m;
    if (row >= rows) row = rows - 1;   // clamp; invalid rows never stored
    const float* __restrict__ wrow = W + (size_t)row * (size_t)cols;

    v8f acc = {};
#pragma unroll 8
    for (int k = 0; k < cols; k += 4) {
        const v2f a = *(const v2f*)(wrow + k + 2 * hi);   // A: W tile
        const v2f b = *(const v2f*)(sx   + k + 2 * hi);   // B: x chunk, all cols
        acc = __builtin_amdgcn_wmma_f32_16x16x4_f32(
                  false, a, false, b, (short)0, acc, false, false);
    }

    // D layout: lanes 0-15 hold rows 0..7 in VGPR 0..7; lanes 16-31 rows 8..15.
    if (m == 0) {
        const int base = row0 + 8 * hi;
#pragma unroll
        for (int i = 0; i < 8; ++i) {
            const int r = base + i;
            if (r < rows) y[r] = acc[i] + bias[r];
        }
    }
}

// ---------------------------------------------------------------------------
// Glue kernels
// ---------------------------------------------------------------------------
__global__ void k_prep(const int* __restrict__ tok,
                       const float* __restrict__ hidden,
                       const float* __restrict__ emb,
                       float* __restrict__ x1, float* __restrict__ x2) {
    const int i = blockIdx.x * blockDim.x + threadIdx.x;
    if (i >= EH_SZ) return;
    if (i < E_SZ) {
        const float v = emb[(size_t)tok[0] * E_SZ + i];
        x1[i] = v;
        x2[i] = v;
    } else {
        x1[i] = hidden[i - E_SZ];
    }
}

__global__ void k_softmax512(const float* __restrict__ s,
                             float* __restrict__ w_ws,
                             float* __restrict__ w_out) {
    __shared__ float red[256];
    const int t = threadIdx.x;
    float m = -3.402823466e38f;
    for (int i = t; i < L_SZ; i += 256) m = fmaxf(m, s[i]);
    red[t] = m; __syncthreads();
    for (int o = 128; o; o >>= 1) { if (t < o) red[t] = fmaxf(red[t], red[t + o]); __syncthreads(); }
    m = red[0]; __syncthreads();
    float sum = 0.f;
    for (int i = t; i < L_SZ; i += 256) sum += expf(s[i] - m);
    red[t] = sum; __syncthreads();
    for (int o = 128; o; o >>= 1) { if (t < o) red[t] += red[t + o]; __syncthreads(); }
    const float inv = 1.f / red[0];
    for (int i = t; i < L_SZ; i += 256) {
        const float v = expf(s[i] - m) * inv;
        w_ws[i] = v;
        w_out[i] = v;
    }
}

__global__ void k_attn_apply(const float* __restrict__ w,
                             const float* __restrict__ enc,
                             float* __restrict__ x2) {
    const int j = blockIdx.x * blockDim.x + threadIdx.x;
    if (j >= H_SZ) return;
    float acc = 0.f;
    for (int l = 0; l < L_SZ; ++l)
        acc = fmaf(w[l], enc[(size_t)l * H_SZ + j], acc);   // coalesced over j
    x2[E_SZ + j] = acc;
}

__global__ void k_relu(const float* __restrict__ in, float* __restrict__ out) {
    const int i = blockIdx.x * blockDim.x + threadIdx.x;
    if (i < H_SZ) out[i] = fmaxf(in[i], 0.f);
}

__global__ void k_gru(const float* __restrict__ gi, const float* __restrict__ gh,
                      const float* __restrict__ h0,
                      float* __restrict__ h1_ws, float* __restrict__ h1_out) {
    const int j = blockIdx.x * blockDim.x + threadIdx.x;
    if (j >= H_SZ) return;
    const float r = 1.f / (1.f + expf(-(gi[j] + gh[j])));
    const float z = 1.f / (1.f + expf(-(gi[H_SZ + j] + gh[H_SZ + j])));
    const float n = tanhf(gi[2 * H_SZ + j] + r * gh[2 * H_SZ + j]);
    const float h = (1.f - z) * n + z * h0[j];
    h1_ws[j]  = h;
    h1_out[j] = h;
}

__global__ void k_lse(const float* __restrict__ x, int n, float* __restrict__ outv) {
    __shared__ float red[256];
    const int t = threadIdx.x;
    float m = -3.402823466e38f;
    for (int i = t; i < n; i += 256) m = fmaxf(m, x[i]);
    red[t] = m; __syncthreads();
    for (int o = 128; o; o >>= 1) { if (t < o) red[t] = fmaxf(red[t], red[t + o]); __syncthreads(); }
    m = red[0]; __syncthreads();
    float s = 0.f;
    for (int i = t; i < n; i += 256) s += expf(x[i] - m);
    red[t] = s; __syncthreads();
    for (int o = 128; o; o >>= 1) { if (t < o) red[t] += red[t + o]; __syncthreads(); }
    if (t == 0) outv[0] = m + logf(red[0]);
}

__global__ void k_logp(const float* __restrict__ logits,
                       const float* __restrict__ lse,
                       float* __restrict__ out, int n) {
    const int i = blockIdx.x * blockDim.x + threadIdx.x;
    if (i < n) out[i] = logits[i] - lse[0];
}

// ---------------------------------------------------------------------------
// Launcher
// ---------------------------------------------------------------------------
extern "C" void kernel_launch(void* const* d_in, const int* in_sizes, int n_in,
                              void* d_out, int out_size, void* d_ws, size_t ws_size,
                              hipStream_t stream) {
    const int*   tok    = (const int*)  d_in[0];
    const float* hidden = (const float*)d_in[1];
    const float* enc    = (const float*)d_in[2];
    const float* emb    = (const float*)d_in[3];
    const float* attn_W = (const float*)d_in[4];
    const float* attn_b = (const float*)d_in[5];
    const float* comb_W = (const float*)d_in[6];
    const float* comb_b = (const float*)d_in[7];
    const float* W_ih   = (const float*)d_in[8];
    const float* W_hh   = (const float*)d_in[9];
    const float* b_ih   = (const float*)d_in[10];
    const float* b_hh   = (const float*)d_in[11];
    const float* out_W  = (const float*)d_in[12];
    const float* out_b  = (const float*)d_in[13];

    float* out = (float*)d_out;   // [logp 50257 | h1 1024 | attn_w 512]
    float* ws  = (float*)d_ws;

    // embedded + cat(embedded, h0)
    k_prep<<<6, 256, 0, stream>>>(tok, hidden, emb, ws + OFF_X1, ws + OFF_X2);

    // attn logits: 512 x 1536 (32 tiles -> 4 blocks of 8 waves)
    wmma_gemv<<<4, 256, EH_SZ * sizeof(float), stream>>>(
        attn_W, ws + OFF_X1, attn_b, ws + OFF_SCORES, L_SZ, EH_SZ);

    // softmax -> attn weights (also written to output slot)
    k_softmax512<<<1, 256, 0, stream>>>(ws + OFF_SCORES, ws + OFF_ATTNW,
                                        out + V_SZ + H_SZ);

    // attn_applied = w @ encoder_outputs  (coalesced column reduction)
    k_attn_apply<<<4, 256, 0, stream>>>(ws + OFF_ATTNW, enc, ws + OFF_X2);

    // combine: 1024 x 1536 (+ relu)
    wmma_gemv<<<8, 256, EH_SZ * sizeof(float), stream>>>(
        comb_W, ws + OFF_X2, comb_b, ws + OFF_PRERELU, H_SZ, EH_SZ);
    k_relu<<<4, 256, 0, stream>>>(ws + OFF_PRERELU, ws + OFF_X3);

    // GRU matmuls: 3072 x 1024 each (192 tiles -> 24 blocks)
    wmma_gemv<<<24, 256, H_SZ * sizeof(float), stream>>>(
        W_ih, ws + OFF_X3, b_ih, ws + OFF_GI, 3 * H_SZ, H_SZ);
    wmma_gemv<<<24, 256, H_SZ * sizeof(float), stream>>>(
        W_hh, hidden, b_hh, ws + OFF_GH, 3 * H_SZ, H_SZ);
    k_gru<<<4, 256, 0, stream>>>(ws + OFF_GI, ws + OFF_GH, hidden,
                                 ws + OFF_H1, out + V_SZ);

    // output projection: 50257 x 1024 -> 3142 tiles -> 393 blocks
    wmma_gemv<<<393, 256, H_SZ * sizeof(float), stream>>>(
        out_W, ws + OFF_H1, out_b, ws + OFF_LOGITS, V_SZ, H_SZ);

    // log_softmax
    k_lse<<<1, 256, 0, stream>>>(ws + OFF_LOGITS, V_SZ, ws + OFF_RED);
    k_logp<<<197, 256, 0, stream>>>(ws + OFF_LOGITS, ws + OFF_RED, out, V_SZ);
}